// GCNSurrogateModel_30657476559342
// MI455X (gfx1250) — compile-verified
//
#include <hip/hip_runtime.h>
#include <math.h>

typedef __attribute__((ext_vector_type(2))) float v2f;
typedef __attribute__((ext_vector_type(8))) float v8f;

#define N_NODES 100000
#define N_EDGES 1600000
#define POOL_BLOCKS 256

// ---------------- degree / normalization ----------------

__global__ void init_deg_kernel(float* deg, int n) {
    int i = blockIdx.x * blockDim.x + threadIdx.x;
    if (i < n) deg[i] = 1.0f;                 // self-loop contributes 1
}

__global__ void count_deg_kernel(const long long* __restrict__ dst, float* deg, int e) {
    int i = blockIdx.x * blockDim.x + threadIdx.x;
    if (i < e) atomicAdd(&deg[(int)dst[i]], 1.0f);
}

__global__ void dinv_kernel(float* deg, int n) {
    int i = blockIdx.x * blockDim.x + threadIdx.x;
    if (i < n) deg[i] = rsqrtf(deg[i]);       // deg >= 1 always (self-loops)
}

__global__ void norm_kernel(const long long* __restrict__ src,
                            const long long* __restrict__ dst,
                            const float* __restrict__ dinv,
                            float* __restrict__ nrm, int e) {
    int i = blockIdx.x * blockDim.x + threadIdx.x;
    if (i < e) nrm[i] = dinv[(int)src[i]] * dinv[(int)dst[i]];
}

// ---------------- WMMA f32 GEMM: H[M,NF] = X[M,K] @ W[K,NF] ----------------
// K, NF compile-time; M = N_NODES = 6250 * 16 exactly, so no row guards.
// One wave computes one 16x16 output tile via V_WMMA_F32_16X16X4_F32.
// A fragment (16x4 f32, 2 VGPRs): lanes 0-15 hold K=k0,k0+1 ; lanes 16-31 K=k0+2,k0+3
// -> 8-byte aligned pair, loaded as one b64.
// B fragment (4x16): mirrored (N striped across lanes).
// C/D: VGPR r -> row = 8*half + r, col = lane&15.

template <int K, int NF>
__global__ __launch_bounds__(32)
void gemm_wmma_f32_kernel(const float* __restrict__ X, const float* __restrict__ W,
                          float* __restrict__ H) {
    const int lane = threadIdx.x & 31;
    const int half = lane >> 4;        // 0 or 1
    const int l16  = lane & 15;
    const int rowA = blockIdx.x * 16 + l16;
    const int colB = blockIdx.y * 16 + l16;

    const float* __restrict__ xrow = X + rowA * K + 2 * half;       // A: this lane's K pair base
    const float* __restrict__ wcol = W + 2 * half * NF + colB;      // B: this lane's K pair base

    v8f c = {0.f, 0.f, 0.f, 0.f, 0.f, 0.f, 0.f, 0.f};

    #pragma unroll
    for (int k0 = 0; k0 < K; k0 += 4) {
        v2f a = *(const v2f*)(xrow + k0);          // 64-bit load: K[k], K[k+1]
        v2f b;
        b.x = wcol[k0 * NF];                        // immediate offsets (NF const)
        b.y = wcol[k0 * NF + NF];
        c = __builtin_amdgcn_wmma_f32_16x16x4_f32(false, a, false, b,
                                                  (short)0, c, false, false);
    }

    float* __restrict__ hrow = H + (blockIdx.x * 16 + half * 8) * NF + blockIdx.y * 16 + l16;
    #pragma unroll
    for (int r = 0; r < 8; ++r) hrow[r * NF] = c[r];
}

// ---------------- aggregation ----------------

// agg[i,f] = h[i,f] * dinv[i]^2   (self-loop term; also initializes agg)
template <int LOGF>
__global__ void selfloop_init_kernel(const float* __restrict__ h,
                                     const float* __restrict__ dinv,
                                     float* __restrict__ agg, int total) {
    int i = blockIdx.x * blockDim.x + threadIdx.x;
    if (i < total) {
        float d = dinv[i >> LOGF];
        agg[i] = h[i] * d * d;
    }
}

// agg[dst] += h[src] * norm[e]   (one thread per edge, float4 gathers, L2 atomics)
template <int F>
__global__ void edge_scatter_kernel(const float* __restrict__ h,
                                    const float* __restrict__ nrm,
                                    const long long* __restrict__ src,
                                    const long long* __restrict__ dst,
                                    float* __restrict__ agg, int e) {
    int i = blockIdx.x * blockDim.x + threadIdx.x;
    if (i >= e) return;
    const int s = (int)src[i];
    const int d = (int)dst[i];
    const float nm = nrm[i];
    const float4* __restrict__ hs = (const float4*)(h + s * F);
    float* __restrict__ ad = agg + d * F;
    #pragma unroll
    for (int f4 = 0; f4 < (F >> 2); ++f4) {
        float4 v = hs[f4];
        atomicAdd(&ad[4 * f4 + 0], v.x * nm);
        atomicAdd(&ad[4 * f4 + 1], v.y * nm);
        atomicAdd(&ad[4 * f4 + 2], v.z * nm);
        atomicAdd(&ad[4 * f4 + 3], v.w * nm);
    }
}

// agg[i,f] = tanh(agg[i,f] + b[f])  (in place; result becomes next layer input)
template <int F>
__global__ void finalize_kernel(float* __restrict__ agg,
                                const float* __restrict__ bias, int total) {
    int i = blockIdx.x * blockDim.x + threadIdx.x;
    if (i < total) {
        agg[i] = tanhf(agg[i] + bias[i & (F - 1)]);
    }
}

// ---------------- max pool + head ----------------

__global__ void pool_partial_kernel(const float* __restrict__ h3,
                                    float* __restrict__ partial, int n) {
    const int f = threadIdx.x;          // 64 features
    const int b = blockIdx.x;           // POOL_BLOCKS blocks
    const int chunk = (n + gridDim.x - 1) / gridDim.x;
    const int r0 = b * chunk;
    int r1 = r0 + chunk; if (r1 > n) r1 = n;
    float m = -INFINITY;
    for (int r = r0; r < r1; ++r) m = fmaxf(m, h3[r * 64 + f]);
    partial[b * 64 + f] = m;
}

__global__ void pool_final_kernel(const float* __restrict__ partial,
                                  float* __restrict__ pooled, int nb) {
    const int f = threadIdx.x;          // 64
    float m = -INFINITY;
    for (int b = 0; b < nb; ++b) m = fmaxf(m, partial[b * 64 + f]);
    pooled[f] = m;
}

__global__ void head_kernel(const float* __restrict__ pooled,
                            const float* __restrict__ Wl,
                            const float* __restrict__ bl,
                            float* __restrict__ out) {
    __shared__ float s[64];
    const int f = threadIdx.x;          // 64
    s[f] = pooled[f] * Wl[f];
    __syncthreads();
    for (int st = 32; st > 0; st >>= 1) {
        if (f < st) s[f] += s[f + st];
        __syncthreads();
    }
    if (f == 0) out[0] = s[0] + bl[0];
}

// ---------------- launch ----------------

extern "C" void kernel_launch(void* const* d_in, const int* in_sizes, int n_in,
                              void* d_out, int out_size, void* d_ws, size_t ws_size,
                              hipStream_t stream) {
    const float* x  = (const float*)d_in[0];
    const float* W1 = (const float*)d_in[1];
    const float* b1 = (const float*)d_in[2];
    const float* W2 = (const float*)d_in[3];
    const float* b2 = (const float*)d_in[4];
    const float* W3 = (const float*)d_in[5];
    const float* b3 = (const float*)d_in[6];
    const float* Wl = (const float*)d_in[7];
    const float* bl = (const float*)d_in[8];
    const long long* ei = (const long long*)d_in[9];   // int64 [2, E]
    const long long* src = ei;
    const long long* dst = ei + N_EDGES;

    float* ws      = (float*)d_ws;
    float* dinv    = ws;                                // N
    float* nrm     = dinv + N_NODES;                    // E
    float* h       = nrm + N_EDGES;                     // N*64 (max)
    float* agg     = h + (size_t)N_NODES * 64;          // N*64 (max)
    float* partial = agg + (size_t)N_NODES * 64;        // POOL_BLOCKS*64
    float* pooled  = partial + POOL_BLOCKS * 64;        // 64

    const int TB = 256;
    const int mtiles = N_NODES / 16;                    // 6250 (exact)

    // graph normalization (computed once, reused by all 3 layers)
    init_deg_kernel<<<(N_NODES + TB - 1) / TB, TB, 0, stream>>>(dinv, N_NODES);
    count_deg_kernel<<<(N_EDGES + TB - 1) / TB, TB, 0, stream>>>(dst, dinv, N_EDGES);
    dinv_kernel<<<(N_NODES + TB - 1) / TB, TB, 0, stream>>>(dinv, N_NODES);
    norm_kernel<<<(N_EDGES + TB - 1) / TB, TB, 0, stream>>>(src, dst, dinv, nrm, N_EDGES);

    // ---- layer 1: 128 -> 16 ----
    gemm_wmma_f32_kernel<128, 16><<<dim3(mtiles, 1), 32, 0, stream>>>(x, W1, h);
    selfloop_init_kernel<4><<<(N_NODES * 16 + TB - 1) / TB, TB, 0, stream>>>(h, dinv, agg, N_NODES * 16);
    edge_scatter_kernel<16><<<(N_EDGES + TB - 1) / TB, TB, 0, stream>>>(h, nrm, src, dst, agg, N_EDGES);
    finalize_kernel<16><<<(N_NODES * 16 + TB - 1) / TB, TB, 0, stream>>>(agg, b1, N_NODES * 16);

    // ---- layer 2: 16 -> 32 (input = agg) ----
    gemm_wmma_f32_kernel<16, 32><<<dim3(mtiles, 2), 32, 0, stream>>>(agg, W2, h);
    selfloop_init_kernel<5><<<(N_NODES * 32 + TB - 1) / TB, TB, 0, stream>>>(h, dinv, agg, N_NODES * 32);
    edge_scatter_kernel<32><<<(N_EDGES + TB - 1) / TB, TB, 0, stream>>>(h, nrm, src, dst, agg, N_EDGES);
    finalize_kernel<32><<<(N_NODES * 32 + TB - 1) / TB, TB, 0, stream>>>(agg, b2, N_NODES * 32);

    // ---- layer 3: 32 -> 64 (input = agg) ----
    gemm_wmma_f32_kernel<32, 64><<<dim3(mtiles, 4), 32, 0, stream>>>(agg, W3, h);
    selfloop_init_kernel<6><<<(N_NODES * 64 + TB - 1) / TB, TB, 0, stream>>>(h, dinv, agg, N_NODES * 64);
    edge_scatter_kernel<64><<<(N_EDGES + TB - 1) / TB, TB, 0, stream>>>(h, nrm, src, dst, agg, N_EDGES);
    finalize_kernel<64><<<(N_NODES * 64 + TB - 1) / TB, TB, 0, stream>>>(agg, b3, N_NODES * 64);

    // ---- global max pool + linear head ----
    pool_partial_kernel<<<POOL_BLOCKS, 64, 0, stream>>>(agg, partial, N_NODES);
    pool_final_kernel<<<1, 64, 0, stream>>>(partial, pooled, POOL_BLOCKS);
    head_kernel<<<1, 64, 0, stream>>>(pooled, Wl, bl, (float*)d_out);
}